// URNNCell_26027501814580
// MI455X (gfx1250) — compile-verified
//
#include <hip/hip_runtime.h>
#include <math.h>

typedef __attribute__((ext_vector_type(2))) float v2f;
typedef __attribute__((ext_vector_type(8))) float v8f;
typedef __attribute__((ext_vector_type(4))) unsigned int v4u;
typedef __attribute__((ext_vector_type(8))) int v8i;
typedef __attribute__((ext_vector_type(4))) int v4i;

#if defined(__HIP_DEVICE_COMPILE__)
#if !__has_builtin(__builtin_amdgcn_wmma_f32_16x16x4_f32)
#error "DEVICE pass: __builtin_amdgcn_wmma_f32_16x16x4_f32 not available"
#endif
#if !__has_builtin(__builtin_amdgcn_tensor_load_to_lds)
#error "DEVICE pass: __builtin_amdgcn_tensor_load_to_lds not available"
#endif
#endif

#define BATCH 8192
#define NN    1024   // N
#define N2    2048   // 2N
#define NIN   256

// LDS W-tile: 64 rows x 128 K dwords, padded to 132-dword stride (TDM pad)
#define KSTAGE   128
#define LDS_PITCH 132   // 128 + 4 pad dwords -> stride mod 64 banks == 4 (conflict-free)

// ---------------------------------------------------------------------------
// Kernel 1: inputs_mul = inputs @ w_ih.T  (8192x256)x(256x2048) -> (8192x2048)
//
// Block = 8 waves computes a 256(M) x 64(N) tile. The shared operand (64 rows
// of W == 64 cols of B) is staged into LDS once per K-half by the Tensor Data
// Mover (TENSOR_LOAD_TO_LDS, wave 0 issues, s_wait_tensorcnt, barrier).
// Each wave owns 32 rows: 2 A fragments x 4 B fragments = 8 f32 accumulators,
// all math in exact f32 via V_WMMA_F32_16X16X4_F32.
//
// Fragment layouts (ISA 7.12.2, 32-bit):
//  A 16x4:  lanes 0-15 -> M=lane,  K=k0,k0+1 ; lanes 16-31 -> M=lane-16, K=k0+2,k0+3
//  B 4x16:  lanes 0-15 -> N=lane,  rows K=k0/k0+1 ; lanes 16-31 -> K=k0+2/k0+3
//  C/D 16x16: VGPR r: lanes 0-15 -> M=r, N=lane ; lanes 16-31 -> M=r+8
// ---------------------------------------------------------------------------
__global__ __launch_bounds__(256) void urnn_gemm(const float* __restrict__ A,
                                                 const float* __restrict__ W,
                                                 float* __restrict__ C)
{
    __shared__ float ldsW[64 * LDS_PITCH];    // 33792 B

    const int lane    = threadIdx.x & 31;
    const int wave    = threadIdx.x >> 5;
    const int lane16  = lane & 15;
    const int halfSel = lane >> 4;            // 0: K pair (k0,k0+1), 1: (k0+2,k0+3)

    const int tm   = blockIdx.x >> 5;         // 0..31  (256-row blocks)
    const int tn   = blockIdx.x & 31;         // 0..31  (64-col blocks)
    const int row0 = tm * 256;
    const int col0 = tn * 64;

    v8f acc[8] = {};                          // [m*4+c], m in {0,1}, c in 0..3

    const float* __restrict__ Arow = A + (size_t)(row0 + wave * 32 + lane16) * NIN;

    const unsigned int ldsOff = (unsigned int)(uintptr_t)&ldsW[0];
    const unsigned long long gbase =
        (unsigned long long)(uintptr_t)(W + (size_t)col0 * NIN);

    for (int s = 0; s < 2; ++s) {
        __syncthreads();                      // prev-stage LDS consumers done
        if (wave == 0) {
            // ---- Tensor DMA descriptor: 2D tile, X=128 dwords of K, Y=64 rows
            const unsigned long long ga = gbase + (unsigned long long)s * (KSTAGE * 4);
            v4u g0;
            g0[0] = 1u;                                   // count=1 (valid D#)
            g0[1] = ldsOff;                               // lds_addr (bytes)
            g0[2] = (unsigned int)(ga & 0xffffffffull);   // global_addr[31:0]
            g0[3] = (unsigned int)((ga >> 32) & 0x01ffffffull) | (2u << 30); // [56:32] | type=2
            v8i g1;
            g1[0] = (int)((2u << 16)      // data_size = 4 bytes
                        | (1u << 20)      // pad_enable
                        | (6u << 22)      // pad_interval: 128 dwords
                        | (3u << 25));    // pad_amount: 4 dwords
            g1[1] = (int)(((unsigned)NIN & 0xffffu) << 16);   // tensor_dim0 lo16 @ [63:48]
            g1[2] = (int)(((unsigned)N2 & 0xffffu) << 16);    // dim0 hi=0; tensor_dim1 lo16 @ [95:80]
            g1[3] = (int)(((unsigned)KSTAGE) << 16);          // dim1 hi=0; tile_dim0=128 @ [127:112]
            g1[4] = (int)(64u);                               // tile_dim1=64; tile_dim2=0
            g1[5] = (int)((unsigned)NIN);                     // tensor_dim0_stride lo32 = 256
            g1[6] = (int)(((unsigned)NIN & 0xffffu) << 16);   // d0s hi=0; tensor_dim1_stride lo16
            g1[7] = 0;                                        // tensor_dim1_stride hi = 0
            v4i g2 = {0, 0, 0, 0};
            v4i g3 = {0, 0, 0, 0};
            v8i g4 = {0, 0, 0, 0, 0, 0, 0, 0};                // unused trailing group (6-arg form)
            __builtin_amdgcn_tensor_load_to_lds(g0, g1, g2, g3, g4, 0);
            __builtin_amdgcn_s_wait_tensorcnt(0);
        }
        __syncthreads();                      // tile visible to all 8 waves

        for (int kk = 0; kk < KSTAGE; kk += 4) {
            const int kl = kk + 2 * halfSel;              // K offset in LDS stage
            const int kg = s * KSTAGE + kl;               // K offset in A
            v2f a0 = *(const v2f*)(Arow + kg);
            v2f a1 = *(const v2f*)(Arow + 16 * NIN + kg);
            v2f b0 = *(const v2f*)&ldsW[(0 * 16 + lane16) * LDS_PITCH + kl];
            v2f b1 = *(const v2f*)&ldsW[(1 * 16 + lane16) * LDS_PITCH + kl];
            v2f b2 = *(const v2f*)&ldsW[(2 * 16 + lane16) * LDS_PITCH + kl];
            v2f b3 = *(const v2f*)&ldsW[(3 * 16 + lane16) * LDS_PITCH + kl];
#if defined(__HIP_DEVICE_COMPILE__)
            acc[0] = __builtin_amdgcn_wmma_f32_16x16x4_f32(false, a0, false, b0, (short)0, acc[0], false, false);
            acc[1] = __builtin_amdgcn_wmma_f32_16x16x4_f32(false, a0, false, b1, (short)0, acc[1], false, false);
            acc[2] = __builtin_amdgcn_wmma_f32_16x16x4_f32(false, a0, false, b2, (short)0, acc[2], false, false);
            acc[3] = __builtin_amdgcn_wmma_f32_16x16x4_f32(false, a0, false, b3, (short)0, acc[3], false, false);
            acc[4] = __builtin_amdgcn_wmma_f32_16x16x4_f32(false, a1, false, b0, (short)0, acc[4], false, false);
            acc[5] = __builtin_amdgcn_wmma_f32_16x16x4_f32(false, a1, false, b1, (short)0, acc[5], false, false);
            acc[6] = __builtin_amdgcn_wmma_f32_16x16x4_f32(false, a1, false, b2, (short)0, acc[6], false, false);
            acc[7] = __builtin_amdgcn_wmma_f32_16x16x4_f32(false, a1, false, b3, (short)0, acc[7], false, false);
#endif
        }
    }

#pragma unroll
    for (int m = 0; m < 2; ++m) {
#pragma unroll
        for (int r = 0; r < 8; ++r) {
            const int row = row0 + wave * 32 + m * 16 + r + 8 * halfSel;
            float* p = C + (size_t)row * N2 + col0 + lane16;
            p[0]  = acc[m * 4 + 0][r];
            p[16] = acc[m * 4 + 1][r];
            p[32] = acc[m * 4 + 2][r];
            p[48] = acc[m * 4 + 3][r];
        }
    }
}

// ---------------------------------------------------------------------------
// Kernel 2: per-row complex chain, one 256-thread block per batch row.
// LDS: zr/zi[1024] (8KB) + 3x256 reduction scratch (3KB).
// ---------------------------------------------------------------------------
__device__ __forceinline__ void fft1024(float* zr, float* zi, int tid, float sign)
{
#pragma unroll
    for (int q = 0; q < 4; ++q) {
        const int i = tid + 256 * q;
        const int j = (int)(__brev((unsigned)i) >> 22);
        if (i < j) {
            float tr = zr[i], ti = zi[i];
            zr[i] = zr[j]; zi[i] = zi[j];
            zr[j] = tr;    zi[j] = ti;
        }
    }
    __syncthreads();

    for (int s = 1; s <= 10; ++s) {
        const int m = 1 << s, half = m >> 1;
#pragma unroll
        for (int q = 0; q < 2; ++q) {
            const int t   = tid + 256 * q;          // 0..511, disjoint butterflies
            const int grp = t >> (s - 1);
            const int pos = t & (half - 1);
            const int i   = grp * m + pos;
            const int j   = i + half;
            const float ang = sign * 6.2831853071795864769f * (float)pos / (float)m;
            float sn, cs;
            __sincosf(ang, &sn, &cs);
            const float xr = zr[j], xi = zi[j];
            const float tr = cs * xr - sn * xi;
            const float ti = cs * xi + sn * xr;
            const float ur = zr[i], ui = zi[i];
            zr[i] = ur + tr; zi[i] = ui + ti;
            zr[j] = ur - tr; zi[j] = ui - ti;
        }
        __syncthreads();
    }
}

__device__ __forceinline__ void reflect1024(float* zr, float* zi,
                                            const float* __restrict__ vre,
                                            const float* __restrict__ vim,
                                            int tid,
                                            float* sr, float* si, float* sn)
{
    float vr[4], vi[4];
    float pr = 0.f, pi = 0.f, pn = 0.f;
#pragma unroll
    for (int q = 0; q < 4; ++q) {
        const int i = tid + 256 * q;
        vr[q] = vre[i]; vi[q] = vim[i];
        const float ar = zr[i], ai = zi[i];
        pr += ar * vr[q] + ai * vi[q];      // Re(z * conj(v))
        pi += ai * vr[q] - ar * vi[q];      // Im(z * conj(v))
        pn += vr[q] * vr[q] + vi[q] * vi[q];
    }
    sr[tid] = pr; si[tid] = pi; sn[tid] = pn;
    __syncthreads();
    for (int off = 128; off > 0; off >>= 1) {
        if (tid < off) {
            sr[tid] += sr[tid + off];
            si[tid] += si[tid + off];
            sn[tid] += sn[tid + off];
        }
        __syncthreads();
    }
    const float f  = 2.0f / sn[0];
    const float cr = f * sr[0];
    const float ci = f * si[0];
    __syncthreads();
#pragma unroll
    for (int q = 0; q < 4; ++q) {
        const int i = tid + 256 * q;
        zr[i] -= cr * vr[q] - ci * vi[q];   // z -= (cr + i*ci) * v
        zi[i] -= cr * vi[q] + ci * vr[q];
    }
    __syncthreads();
}

__global__ __launch_bounds__(256) void urnn_row(
    float* __restrict__ out,                 // inputs_mul on entry, new_state on exit
    const float* __restrict__ state,
    const float* __restrict__ b_h,
    const float* __restrict__ w1,
    const float* __restrict__ w2,
    const float* __restrict__ w3,
    const float* __restrict__ v1re, const float* __restrict__ v1im,
    const float* __restrict__ v2re, const float* __restrict__ v2im,
    const int*   __restrict__ perm)
{
    __shared__ float zr[NN];
    __shared__ float zi[NN];
    __shared__ float sr[256], si[256], sn[256];

    const int tid = threadIdx.x;
    const int b   = blockIdx.x;
    float* __restrict__ orow = out + (size_t)b * N2;
    const float* __restrict__ srow = state + (size_t)b * N2;

    // inputs_mul row -> registers before we overwrite it at the end
    float ir[4], ii[4];
#pragma unroll
    for (int q = 0; q < 4; ++q) {
        const int i = tid + 256 * q;
        ir[q] = orow[i];
        ii[q] = orow[i + NN];
    }

    // z = diag(e^{i w1}) * state_c
#pragma unroll
    for (int q = 0; q < 4; ++q) {
        const int i = tid + 256 * q;
        const float a = srow[i], bm = srow[i + NN];
        float s, c; __sincosf(w1[i], &s, &c);
        zr[i] = c * a - s * bm;
        zi[i] = c * bm + s * a;
    }
    __syncthreads();

    fft1024(zr, zi, tid, -1.0f);                 // forward FFT
    reflect1024(zr, zi, v1re, v1im, tid, sr, si, sn);

    // permutation gather + diag(e^{i w2})
    float gr[4], gi[4];
#pragma unroll
    for (int q = 0; q < 4; ++q) {
        const int i = tid + 256 * q;
        const int p = perm[i];
        float s, c; __sincosf(w2[i], &s, &c);
        const float ar = zr[p], ai = zi[p];
        gr[q] = c * ar - s * ai;
        gi[q] = c * ai + s * ar;
    }
    __syncthreads();
#pragma unroll
    for (int q = 0; q < 4; ++q) {
        const int i = tid + 256 * q;
        zr[i] = gr[q]; zi[i] = gi[q];
    }
    __syncthreads();

    fft1024(zr, zi, tid, +1.0f);                 // inverse FFT (unscaled)
    const float invN = 1.0f / (float)NN;
#pragma unroll
    for (int q = 0; q < 4; ++q) {
        const int i = tid + 256 * q;
        zr[i] *= invN; zi[i] *= invN;
    }
    __syncthreads();

    reflect1024(zr, zi, v2re, v2im, tid, sr, si, sn);

    // diag(e^{i w3}), add inputs_c, modReLU, store
#pragma unroll
    for (int q = 0; q < 4; ++q) {
        const int i = tid + 256 * q;
        float s, c; __sincosf(w3[i], &s, &c);
        const float ar = zr[i], ai = zi[i];
        const float tr = c * ar - s * ai + ir[q];
        const float ti = c * ai + s * ar + ii[q];
        const float norm = sqrtf(tr * tr + ti * ti);
        const float sc = fmaxf(norm + b_h[i], 0.0f) / (norm + 1e-6f);
        orow[i]      = tr * sc;
        orow[i + NN] = ti * sc;
    }
}

// ---------------------------------------------------------------------------
extern "C" void kernel_launch(void* const* d_in, const int* in_sizes, int n_in,
                              void* d_out, int out_size, void* d_ws, size_t ws_size,
                              hipStream_t stream)
{
    const float* inputs = (const float*)d_in[0];
    const float* state  = (const float*)d_in[1];
    const float* w_ih   = (const float*)d_in[2];
    const float* b_h    = (const float*)d_in[3];
    const float* w1     = (const float*)d_in[4];
    const float* w2     = (const float*)d_in[5];
    const float* w3     = (const float*)d_in[6];
    const float* v1re   = (const float*)d_in[7];
    const float* v1im   = (const float*)d_in[8];
    const float* v2re   = (const float*)d_in[9];
    const float* v2im   = (const float*)d_in[10];
    const int*   perm   = (const int*)d_in[11];
    float* out = (float*)d_out;

    // GEMM: 32 row-blocks x 32 col-blocks, 8 waves each (256x64 tile/block)
    urnn_gemm<<<1024, 256, 0, stream>>>(inputs, w_ih, out);
    // Complex chain: one block per batch row
    urnn_row<<<BATCH, 256, 0, stream>>>(out, state, b_h, w1, w2, w3,
                                        v1re, v1im, v2re, v2im, perm);
}